// VectorSpinModel_29609504539239
// MI455X (gfx1250) — compile-verified
//
#include <hip/hip_runtime.h>
#include <hip/hip_bf16.h>
#include <math.h>

// ---------------------------------------------------------------------------
// VectorSpinModel on MI455X (gfx1250), reformulated via eigendecomposition:
//   J = Q L Q^T  (parallel Jacobi eigensolver, A in 256KB LDS of 320KB WGP pool)
//   w[b,i] = sum_f (Q^T h_b)_{i,f}^2   (fp32 WMMA V_WMMA_F32_16X16X4_F32)
//   Newton on t per batch using 1-D eigen sums; then phi / afe.
// ---------------------------------------------------------------------------

typedef float v2f __attribute__((ext_vector_type(2)));
typedef float v8f __attribute__((ext_vector_type(8)));

#define NSPIN 256
#define FDIM  64
#define NBATCH 64
#define NSWEEP 12
#define EIG_THREADS 1024           // 32 waves: 8 threads per rotation pair
#define SUBW 8                     // threads per pair
#define COLS_PER_THREAD (NSPIN / SUBW / 2 * 2 / 2)  // 256/8/2*2/2 = 16

// ---------------------------------------------------------------------------
// Kernel 1: J = 0.5*(Jr + Jr^T) with zero diagonal; QT = I.
// grid: 256 blocks (row i), 256 threads (col j)
// ---------------------------------------------------------------------------
__global__ __launch_bounds__(256)
void build_J_kernel(const float* __restrict__ Jr,
                    float* __restrict__ Jout,
                    float* __restrict__ QT) {
  const int i = blockIdx.x;
  const int j = threadIdx.x;
  const float v = (i == j) ? 0.0f
                           : 0.5f * (Jr[i * NSPIN + j] + Jr[j * NSPIN + i]);
  Jout[i * NSPIN + j] = v;
  QT[i * NSPIN + j] = (i == j) ? 1.0f : 0.0f;
}

// ---------------------------------------------------------------------------
// Kernel 2: parallel cyclic Jacobi eigensolver, single workgroup, 1024 threads.
//   A (256x256 fp32 = 256KB) in dynamic LDS; eigenvectors accumulated as
//   QT (rows of Q^T) in global (resident in the 192MB L2).
//   pair pr = tid>>3 (128 disjoint rotations per round); the 8 threads of a
//   pair are consecutive lanes of one wave32, so their redundant reads of
//   a_pp/a_qq/a_pq complete (lockstep) before any same-pair writes.
//   Column index staggered by pr to spread the 64 LDS banks.
// ---------------------------------------------------------------------------
__global__ __launch_bounds__(EIG_THREADS)
void jacobi_eig_kernel(const float* __restrict__ Jin,
                       float* __restrict__ QT,
                       float* __restrict__ lambda_out) {
  extern __shared__ float A[]; // 256*256 floats
  const int tid = threadIdx.x;
  for (int idx = tid; idx < NSPIN * NSPIN; idx += EIG_THREADS) A[idx] = Jin[idx];
  __syncthreads();

  const int pr = tid >> 3;          // rotation pair 0..127
  const int sub = tid & (SUBW - 1); // 0..7, each owns 32 columns

  for (int sweep = 0; sweep < NSWEEP; ++sweep) {
    for (int r = 0; r < NSPIN - 1; ++r) {
      // round-robin tournament pairing: index 255 fixed, others rotate mod 255
      int p, q;
      if (pr == 0) {
        p = NSPIN - 1;
        q = r;
      } else {
        p = (r + pr) % (NSPIN - 1);
        q = (r + (NSPIN - 1) - pr) % (NSPIN - 1);
      }
      const float app = A[p * NSPIN + p];
      const float aqq = A[q * NSPIN + q];
      const float apq = A[p * NSPIN + q];
      float c = 1.0f, s = 0.0f;
      if (fabsf(apq) > 1e-30f) {
        const float tau = (aqq - app) / (2.0f * apq);
        const float tt = copysignf(1.0f, tau) /
                         (fabsf(tau) + sqrtf(1.0f + tau * tau));
        c = 1.0f / sqrtf(1.0f + tt * tt);
        s = tt * c;
      }
      // Phase A: rows p,q of A (J^T * A), and rows p,q of QT (J^T * QT).
      for (int kk = 0; kk < 32; ++kk) {
        const int j = sub * 32 + ((kk + pr) & 31);
        const float ap = A[p * NSPIN + j];
        const float aq = A[q * NSPIN + j];
        A[p * NSPIN + j] = c * ap - s * aq;
        A[q * NSPIN + j] = s * ap + c * aq;
        const float qp = QT[p * NSPIN + j];
        const float qq = QT[q * NSPIN + j];
        QT[p * NSPIN + j] = c * qp - s * qq;
        QT[q * NSPIN + j] = s * qp + c * qq;
      }
      __syncthreads();
      // Phase B: columns p,q of A ((J^T A) * J)
      for (int kk = 0; kk < 32; ++kk) {
        const int i = sub * 32 + ((kk + pr) & 31);
        const float ap = A[i * NSPIN + p];
        const float aq = A[i * NSPIN + q];
        A[i * NSPIN + p] = c * ap - s * aq;
        A[i * NSPIN + q] = s * ap + c * aq;
      }
      __syncthreads();
    }
  }
  if (tid < NSPIN) lambda_out[tid] = A[tid * NSPIN + tid];
}

// ---------------------------------------------------------------------------
// Kernel 3: G = QT @ h_b via fp32 WMMA (V_WMMA_F32_16X16X4_F32), then
//   w[b,i] = sum_f G[i,f]^2 (G tile never leaves the WGP).
// grid: 64 batches * 16 row-tiles; block: 128 threads = 4 waves,
//   wave wv handles f-tile wv (FDIM=64 -> 4 tiles of 16).
// A-frag layout (16x4 f32): lane L holds A[L&15][k0], A[L&15][k0+1], k0=(L>>4)*2
// B-frag layout (4x16 f32): lane L holds B[k0][L&15], B[k0+1][L&15]
// D layout (16x16 f32):     lane L VGPR r holds D[r + 8*(L>>4)][L&15]
// ---------------------------------------------------------------------------
__global__ __launch_bounds__(128)
void qh_weights_kernel(const float* __restrict__ x,
                       const float* __restrict__ QT,
                       float* __restrict__ w) {
  __shared__ float Gl[4 * 256];
  const int b = blockIdx.x >> 4;
  const int ti = blockIdx.x & 15;
  const int wave = threadIdx.x >> 5;
  const int lane = threadIdx.x & 31;
  const int m = lane & 15;
  const int kk = (lane >> 4) << 1;

  const float* __restrict__ hb = x + (size_t)b * NSPIN * FDIM;
  const int row = ti * 16 + m;   // i index (rows of QT)
  const int f = wave * 16 + m;   // f index (cols of h_b)

  v8f acc = {};
  for (int k = 0; k < NSPIN; k += 4) {
    // A = QT tile: two contiguous fp32 -> single b64 load
    const v2f a = *(const v2f*)(QT + row * NSPIN + k + kk);
    v2f bf;
    bf.x = hb[(k + kk) * FDIM + f];
    bf.y = hb[(k + kk + 1) * FDIM + f];
    acc = __builtin_amdgcn_wmma_f32_16x16x4_f32(
        /*neg_a=*/false, a, /*neg_b=*/false, bf,
        /*c_mod=*/(short)0, acc, /*reuse_a=*/false, /*reuse_b=*/false);
  }

#pragma unroll
  for (int r = 0; r < 8; ++r) {
    const int mm = r + 8 * (lane >> 4);
    Gl[wave * 256 + mm * 16 + (lane & 15)] = acc[r];
  }
  __syncthreads();

  if (threadIdx.x < 16) {
    const int i = threadIdx.x;
    float sum = 0.0f;
#pragma unroll
    for (int wv = 0; wv < 4; ++wv)
      for (int n = 0; n < 16; ++n) {
        const float g = Gl[wv * 256 + i * 16 + n];
        sum = fmaf(g, g, sum);
      }
    w[b * NSPIN + ti * 16 + i] = sum;
  }
}

// ---------------------------------------------------------------------------
// Kernel 4: per-batch Newton on t (40 iters) + phi + afe, all via eigen sums.
//   g(t)  = beta*N - 0.5*sum 1/(t-l) - beta/(4D) * sum w/(t-l)^2
//   g'(t) = 0.5*sum 1/(t-l)^2 + beta/(2D) * sum w/(t-l)^3
//   phi   = beta*N*t - 0.5*sum log|t-l| + beta/(4D)*sum w/(t-l)
//   afe   = -(0.5*log(pi/beta) + phi/N)          (beta = 1)
// grid: 64 blocks (batch), 256 threads (eigen index)
// ---------------------------------------------------------------------------
__global__ __launch_bounds__(256)
void newton_phi_kernel(const float* __restrict__ lam,
                       const float* __restrict__ w,
                       float* __restrict__ out) {
  __shared__ float r1[256], r2[256], r3[256], r4[256];
  __shared__ float tsh;
  const int b = blockIdx.x;
  const int i = threadIdx.x;
  const float li = lam[i];
  const float wi = w[b * NSPIN + i];
  if (i == 0) tsh = 1.0f;
  __syncthreads();

  for (int it = 0; it < 40; ++it) {
    const float t = tsh;
    const float d = t - li;
    const float inv = 1.0f / d;
    const float inv2 = inv * inv;
    r1[i] = inv;
    r2[i] = inv2;
    r3[i] = wi * inv2;
    r4[i] = wi * inv2 * inv;
    __syncthreads();
    for (int sft = 128; sft > 0; sft >>= 1) {
      if (i < sft) {
        r1[i] += r1[i + sft];
        r2[i] += r2[i + sft];
        r3[i] += r3[i + sft];
        r4[i] += r4[i + sft];
      }
      __syncthreads();
    }
    if (i == 0) {
      const float g = 256.0f - 0.5f * r1[0] - (1.0f / 256.0f) * r3[0];
      const float dg = 0.5f * r2[0] + (1.0f / 128.0f) * r4[0];
      const float upd = (fabsf(g) > 1e-4f) ? (g / dg) : 0.0f;
      tsh = t - upd;
    }
    __syncthreads();
  }

  const float t = tsh;
  const float d = t - li;
  r1[i] = logf(fabsf(d));
  r3[i] = wi / d;
  __syncthreads();
  for (int sft = 128; sft > 0; sft >>= 1) {
    if (i < sft) {
      r1[i] += r1[i + sft];
      r3[i] += r3[i + sft];
    }
    __syncthreads();
  }
  if (i == 0) {
    const float phi = 256.0f * t - 0.5f * r1[0] + (1.0f / 256.0f) * r3[0];
    const float afe = -(0.5f * logf(3.14159265358979323846f) + phi / 256.0f);
    out[b] = afe;          // afe, shape (64,)
    out[NBATCH + b] = t;   // t_star, shape (64,1)
  }
}

// ---------------------------------------------------------------------------
// Launch: x = d_in[0] (64*256*64 f32), J_raw = d_in[1] (256*256 f32)
// d_out: 128 f32 (afe[64] ++ t_star[64])
// Workspace (floats): Jbuf[65536] | QT[65536] | lambda[256] | w[64*256]
// ---------------------------------------------------------------------------
extern "C" void kernel_launch(void* const* d_in, const int* in_sizes, int n_in,
                              void* d_out, int out_size, void* d_ws, size_t ws_size,
                              hipStream_t stream) {
  const float* x = (const float*)d_in[0];
  const float* Jraw = (const float*)d_in[1];
  float* ws = (float*)d_ws;

  float* Jbuf = ws;                       // 65536
  float* QT = ws + 65536;                 // 65536
  float* lam = ws + 131072;               // 256
  float* w = ws + 131328;                 // 16384

  build_J_kernel<<<NSPIN, NSPIN, 0, stream>>>(Jraw, Jbuf, QT);
  jacobi_eig_kernel<<<1, EIG_THREADS, NSPIN * NSPIN * sizeof(float), stream>>>(
      Jbuf, QT, lam);
  qh_weights_kernel<<<NBATCH * 16, 128, 0, stream>>>(x, QT, w);
  newton_phi_kernel<<<NBATCH, NSPIN, 0, stream>>>(lam, w, (float*)d_out);
}